// Multi_Frame_Flow_Consistency_Loss_2499670966831
// MI455X (gfx1250) — compile-verified
//
#include <hip/hip_runtime.h>
#include <hip/hip_bf16.h>

#define HD 480
#define WD 640
#define BD 16
#define HWD (HD*WD)
#define PPT 4
#define TPB 256
#define GX (HWD/(TPB*PPT))   // 300 blocks per (batch, direction)

typedef float v2f __attribute__((ext_vector_type(2)));
typedef float v8f __attribute__((ext_vector_type(8)));

// ---------------------------------------------------------------------------
// Setup: fold K, Ki, R, t into per-(direction,batch) affine constants.
//   d1  = depth * (P·[u,v,1]) + CZ            (z-row of K·G·Ki, G = R_dst·R_srcᵀ)
//   uv0 = (pd·(Q·[x,y,1]) + c')_{xy} / (relu(z)+1e-12),  Q = K·Gᵀ·Ki
// 32 threads = 2 directions × 16 batches.
// ---------------------------------------------------------------------------
__global__ void mffc_setup(const float* __restrict__ R0, const float* __restrict__ t0,
                           const float* __restrict__ R1, const float* __restrict__ t1,
                           const float* __restrict__ K,  const float* __restrict__ Ki,
                           float* __restrict__ consts)
{
  int tid = threadIdx.x;
  if (tid >= 32) return;
  int dir = tid >> 4;
  int b   = tid & 15;
  const float* Ra = (dir ? R1 : R0) + b*9;   // source frame rotation
  const float* Rb = (dir ? R0 : R1) + b*9;   // destination frame rotation
  const float* ta = (dir ? t1 : t0) + b*3;
  const float* tb = (dir ? t0 : t1) + b*3;

  float G[3][3];                              // G = Rb · Raᵀ
  for (int i=0;i<3;i++)
    for (int j=0;j<3;j++)
      G[i][j] = Rb[i*3+0]*Ra[j*3+0] + Rb[i*3+1]*Ra[j*3+1] + Rb[i*3+2]*Ra[j*3+2];

  float* o = consts + tid*16;
  // depth row: P = row2(G)·Ki   (K's last row is [0,0,1])
  for (int j=0;j<3;j++)
    o[j] = G[2][0]*Ki[0*3+j] + G[2][1]*Ki[1*3+j] + G[2][2]*Ki[2*3+j];
  o[3] = tb[2] - (G[2][0]*ta[0] + G[2][1]*ta[1] + G[2][2]*ta[2]);

  // T = K·Gᵀ ; Q = T·Ki ; c' = K·ta − T·tb
  float T[3][3];
  for (int i=0;i<3;i++)
    for (int j=0;j<3;j++)
      T[i][j] = K[i*3+0]*G[j][0] + K[i*3+1]*G[j][1] + K[i*3+2]*G[j][2];
  for (int i=0;i<3;i++)
    for (int j=0;j<3;j++)
      o[4 + i*3 + j] = T[i][0]*Ki[0*3+j] + T[i][1]*Ki[1*3+j] + T[i][2]*Ki[2*3+j];
  for (int i=0;i<3;i++)
    o[13+i] = (K[i*3+0]*ta[0]+K[i*3+1]*ta[1]+K[i*3+2]*ta[2])
            - (T[i][0]*tb[0]+T[i][1]*tb[1]+T[i][2]*tb[2]);
}

// ---------------------------------------------------------------------------
// Main fused kernel: one pass over all pixels of one (direction, batch) slab.
// Streams 4 src floats/pixel coalesced; 4-tap bilinear gathers from 5 dst
// fields (uv0 computed on the fly from primary depth — never materialized).
// Deterministic per-block partial (num, den) written to workspace.
// ---------------------------------------------------------------------------
__global__ __launch_bounds__(TPB) void mffc_main(
    const float* __restrict__ depth0, const float* __restrict__ depth1,
    const float* __restrict__ flow0,  const float* __restrict__ flow1,
    const float* __restrict__ amb0,   const float* __restrict__ amb1,
    const float* __restrict__ pd0,    const float* __restrict__ pd1,
    const float* __restrict__ consts, float2* __restrict__ partials)
{
  const int dir = blockIdx.z;
  const int b   = blockIdx.y;

  const float* dsrc = dir ? depth1 : depth0;
  const float* ddst = dir ? depth0 : depth1;
  const float* fsrc = dir ? flow1  : flow0;
  const float* fdst = dir ? flow0  : flow1;
  const float* asrc = dir ? amb1   : amb0;
  const float* adst = dir ? amb0   : amb1;
  const float* pdst = dir ? pd0    : pd1;

  const float* cc = consts + (dir*16 + b)*16;   // wave-uniform -> scalar loads
  const float P0=cc[0], P1=cc[1], P2=cc[2], CZ=cc[3];
  const float Q0=cc[4],Q1=cc[5],Q2=cc[6],Q3=cc[7],Q4=cc[8],Q5=cc[9],
              Q6=cc[10],Q7=cc[11],Q8=cc[12];
  const float CX=cc[13], CY=cc[14], CW=cc[15];

  const int b1 = b*HWD;
  const int b2 = b*2*HWD;

  float num = 0.f, den = 0.f;
  const int base = blockIdx.x * (TPB*PPT);

  #pragma unroll
  for (int k=0;k<PPT;k++) {
    int p = base + k*TPB + (int)threadIdx.x;
    float u = (float)(p % WD);
    float v = (float)(p / WD);
    float d  = dsrc[b1 + p];
    float fx = fsrc[b2 + p];
    float fy = fsrc[b2 + HWD + p];
    float a0 = asrc[b1 + p];

    // depth of src pixel reprojected into dst view
    float d1 = d * (P0*u + P1*v + P2) + CZ;

    // sample position in dst image (grid algebra collapses to u+flow)
    float x = u + fx, y = v + fy;
    float x0f = floorf(x), y0f = floorf(y);
    int   x0 = (int)x0f,   y0 = (int)y0f;
    float wx = x - x0f,    wy = y - y0f;

    float depth10=0.f, f10x=0.f, f10y=0.f, a10=0.f, wu=0.f, wv=0.f;

    #pragma unroll
    for (int t=0;t<4;t++) {
      int xi = x0 + (t & 1);
      int yi = y0 + (t >> 1);
      float wt = ((t&1) ? wx : 1.f-wx) * ((t>>1) ? wy : 1.f-wy);
      if (xi>=0 && xi<WD && yi>=0 && yi<HD) {      // zeros padding
        int idx = yi*WD + xi;
        depth10 += wt * ddst[b1+idx];
        f10x    += wt * fdst[b2+idx];
        f10y    += wt * fdst[b2+HWD+idx];
        a10     += wt * adst[b1+idx];
        // uv0 field value at this integer dst pixel, computed in-place
        float pd = pdst[b1+idx];
        float xf = (float)xi, yf = (float)yi;
        float uz  = pd * (Q6*xf + Q7*yf + Q8) + CW;
        float inv = 1.f / (fmaxf(uz, 0.f) + 1e-12f);
        wu += wt * ((pd * (Q0*xf + Q1*yf + Q2) + CX) * inv);
        wv += wt * ((pd * (Q3*xf + Q4*yf + Q5) + CY) * inv);
      }
    }

    // masks
    float sfx = fx + f10x, sfy = fy + f10y;
    bool fb = (sfx*sfx + sfy*sfy) <
              (0.5f + 0.02f*((fx*fx + fy*fy) + (f10x*f10x + f10y*f10y)));
    bool vc = fabsf(a0 - a10) < 0.01f;
    float ru = wu - u, rv = wv - v;
    bool rf = (ru*ru + rv*rv) < 1.0f;
    if (fb && vc && rf) {
      num += fabsf(d1 - depth10);
      den += 1.f;
    }
  }

  // deterministic in-block reduction: wave32 shuffle tree, then LDS across waves
  for (int off=16; off>0; off>>=1) {
    num += __shfl_xor(num, off, 32);
    den += __shfl_xor(den, off, 32);
  }
  __shared__ float2 wpart[TPB/32];
  int wave = threadIdx.x >> 5;
  if ((threadIdx.x & 31) == 0) wpart[wave] = make_float2(num, den);
  __syncthreads();
  if (threadIdx.x == 0) {
    float n=0.f, dd=0.f;
    #pragma unroll
    for (int i=0;i<TPB/32;i++){ n += wpart[i].x; dd += wpart[i].y; }
    partials[(dir*BD + b)*GX + blockIdx.x] = make_float2(n, dd);
  }
}

// ---------------------------------------------------------------------------
// Final: single wave32. Per-lane serial sums, then V_WMMA_F32_16X16X4_F32
// cross-lane reduction: A = {lane value, 0}, B = all-ones => D[m][*] row-sum
// = v_m + v_{m+16}. Summing the 8 D regs + lanes {0,16} gives the wave total.
// Fixed hardware reduction order -> deterministic across replays.
// ---------------------------------------------------------------------------
__device__ inline float wmma_wave_sum(float s) {
  v2f a;  a[0]  = s;   a[1]  = 0.f;
  v2f bb; bb[0] = 1.f; bb[1] = 1.f;
  v8f c = {};
  v8f dD = __builtin_amdgcn_wmma_f32_16x16x4_f32(
      /*neg_a=*/false, a, /*neg_b=*/false, bb,
      /*c_mod=*/(short)0, c, /*reuse_a=*/false, /*reuse_b=*/false);
  float t = dD[0]+dD[1]+dD[2]+dD[3]+dD[4]+dD[5]+dD[6]+dD[7];
  return __shfl(t, 0, 32) + __shfl(t, 16, 32);
}

__global__ void mffc_final(const float2* __restrict__ partials, float* __restrict__ out) {
  const int lane = threadIdx.x;
  const int nPerDir = BD*GX;         // 4800 = 150*32, no tail divergence
  float loss = 0.f;
  for (int dir=0; dir<2; ++dir) {
    float sn=0.f, sd=0.f;
    for (int i=lane; i<nPerDir; i+=32) {
      float2 p = partials[dir*nPerDir + i];
      sn += p.x; sd += p.y;
    }
    float N = wmma_wave_sum(sn);
    float D = wmma_wave_sum(sd);
    loss += N / (D + 1e-8f);
  }
  if (lane == 0) out[0] = loss;
}

// ---------------------------------------------------------------------------
extern "C" void kernel_launch(void* const* d_in, const int* in_sizes, int n_in,
                              void* d_out, int out_size, void* d_ws, size_t ws_size,
                              hipStream_t stream) {
  const float* depth0 = (const float*)d_in[0];
  const float* depth1 = (const float*)d_in[1];
  const float* R0     = (const float*)d_in[2];
  const float* t0     = (const float*)d_in[3];
  const float* R1     = (const float*)d_in[4];
  const float* t1     = (const float*)d_in[5];
  const float* flow0  = (const float*)d_in[6];
  const float* flow1  = (const float*)d_in[7];
  const float* amb0   = (const float*)d_in[8];
  const float* amb1   = (const float*)d_in[9];
  const float* pd0    = (const float*)d_in[10];
  const float* pd1    = (const float*)d_in[11];
  const float* K      = (const float*)d_in[12];
  const float* Ki     = (const float*)d_in[13];

  float*  consts   = (float*)d_ws;                        // 512 floats used
  float2* partials = (float2*)((char*)d_ws + 4096);       // 2*16*300 float2 = 76.8 KB

  mffc_setup<<<1, 32, 0, stream>>>(R0, t0, R1, t1, K, Ki, consts);
  mffc_main<<<dim3(GX, BD, 2), TPB, 0, stream>>>(
      depth0, depth1, flow0, flow1, amb0, amb1, pd0, pd1, consts, partials);
  mffc_final<<<1, 32, 0, stream>>>(partials, (float*)d_out);
}